// pRotatEScore_76124000354701
// MI455X (gfx1250) — compile-verified
//
#include <hip/hip_runtime.h>
#include <hip/hip_bf16.h>

typedef float v2f __attribute__((ext_vector_type(2)));
typedef float v8f __attribute__((ext_vector_type(8)));

#define GAMMA_F    12.0f
#define EMB_INIT_F 0.109375f
#define DIM        128

// One wave32 handles 16 edges. Lanes l (<16) and l+16 cooperate on edge l:
// per 4-dim K-chunk, half 0 loads dims {k,k+1}, half 1 loads dims {k+2,k+3}
// (exactly the B-matrix layout of V_WMMA_F32_16X16X4_F32). A = all-ones, so
// D[i][j] accumulates the full 128-dim sum of |sin(phase)| for edge j; the
// result is read from acc[0] (M=0 row) in lanes 0..15 (N = lane).
__global__ __launch_bounds__(256)
void pRotatEScore_76124000354701_kernel(const float* __restrict__ node_emb,
                                        const float* __restrict__ rel_emb,
                                        const float* __restrict__ modulus,
                                        const int*   __restrict__ src_idx,
                                        const int*   __restrict__ dst_idx,
                                        const int*   __restrict__ rel_idx,
                                        float*       __restrict__ out,
                                        int n_edges)
{
    const int lane    = threadIdx.x & 31;
    const int wave_ib = threadIdx.x >> 5;
    const long wave   = (long)blockIdx.x * 8 + wave_ib;
    const long base   = wave * 16;
    if (base >= n_edges) return;            // wave-uniform exit; EXEC stays full

    const int e_local = lane & 15;
    const int half    = lane >> 4;

    long edge = base + e_local;
    if (edge >= n_edges) edge = n_edges - 1; // clamp loads; keep EXEC all-1s for WMMA

    const int si = src_idx[edge];
    const int di = dst_idx[edge];
    const int ri = rel_idx[edge];

    const float* __restrict__ ph = node_emb + (long)si * DIM;
    const float* __restrict__ pt = node_emb + (long)di * DIM;
    const float* __restrict__ pr = rel_emb  + (long)ri * DIM;

    // sin(x * pi/emb_init) with hw v_sin (arg in revolutions): x / (2*emb_init)
    const float norm = 1.0f / (2.0f * EMB_INIT_F);

    v2f ones; ones.x = 1.0f; ones.y = 1.0f;  // A-matrix: 16x4 of 1.0
    v8f acc = {};                            // C/D accumulator (16x16 f32)

    #pragma unroll 4
    for (int k = 0; k < DIM; k += 4) {
        const int off = k + 2 * half;
        v2f h = *(const v2f*)(ph + off);
        v2f t = *(const v2f*)(pt + off);
        v2f r = *(const v2f*)(pr + off);
        v2f b;
        b.x = __builtin_fabsf(__builtin_amdgcn_sinf((h.x + r.x - t.x) * norm));
        b.y = __builtin_fabsf(__builtin_amdgcn_sinf((h.y + r.y - t.y) * norm));
        // D = A(ones) x B(|sin| chunk) + C : accumulates column sums (per-edge)
        acc = __builtin_amdgcn_wmma_f32_16x16x4_f32(
                  /*neg_a=*/false, ones, /*neg_b=*/false, b,
                  /*c_mod=*/(short)0, acc, /*reuse_a=*/false, /*reuse_b=*/false);
    }

    if (half == 0) {
        const long eo = base + e_local;
        if (eo < n_edges) {
            out[eo] = (GAMMA_F - acc[0]) * modulus[0];
        }
    }
}

extern "C" void kernel_launch(void* const* d_in, const int* in_sizes, int n_in,
                              void* d_out, int out_size, void* d_ws, size_t ws_size,
                              hipStream_t stream) {
    const float* node_emb = (const float*)d_in[0];
    const float* rel_emb  = (const float*)d_in[1];
    const float* modulus  = (const float*)d_in[2];
    const int*   src_idx  = (const int*)d_in[3];
    const int*   dst_idx  = (const int*)d_in[4];
    const int*   rel_idx  = (const int*)d_in[5];
    float*       out      = (float*)d_out;

    const int E      = in_sizes[3];
    const int waves  = (E + 15) / 16;
    const int blocks = (waves + 7) / 8;     // 8 waves (256 threads) per block

    pRotatEScore_76124000354701_kernel<<<blocks, 256, 0, stream>>>(
        node_emb, rel_emb, modulus, src_idx, dst_idx, rel_idx, out, E);
}